// GIN_65283502899905
// MI455X (gfx1250) — compile-verified
//
#include <hip/hip_runtime.h>
#include <hip/hip_bf16.h>

// ---------------------------------------------------------------------------
// GIN layer for MI455X (gfx1250, wave32, WMMA).
//   agg = segment_sum(x[src]*w, dst); h = (1+eps)x + agg
//   h1 = h @ w1 + b1 ; BN(batch stats) ; ReLU ; out = h1 @ w2 + b2
// fp32 end-to-end using V_WMMA_F32_16X16X4_F32 (memory-bound workload:
// ~26 FLOP/B -> fp32 matrix pipe is the right precision; the win is in
// data movement, so each block now computes a 32-row M tile so every B
// fragment load feeds TWO WMMAs).
// ---------------------------------------------------------------------------

#define GIN_N      100000
#define GIN_E      1600000
#define D_IN       128
#define D_HID      512
#define D_OUT      256
#define EPS1       1e-9f
#define BN_EPS     1e-5f
#define KC         128        // K chunk staged in LDS per iteration

typedef __attribute__((ext_vector_type(2))) float v2f;
typedef __attribute__((ext_vector_type(8))) float v8f;

// ---------------------------------------------------------------------------
// Workspace layout (floats):  agg | sum | sumsq | scale | shift | h1
// ---------------------------------------------------------------------------
#define WS_AGG_ELEMS   ((size_t)GIN_N * D_IN)      // 12.8M

// ---------------------------------------------------------------------------
__global__ __launch_bounds__(256) void zero_f32(float* __restrict__ p, size_t n) {
    size_t i = (size_t)blockIdx.x * blockDim.x + threadIdx.x;
    size_t stride = (size_t)gridDim.x * blockDim.x;
    for (; i < n; i += stride) p[i] = 0.0f;
}

// ---------------------------------------------------------------------------
// One thread per (edge, 4 features): L2-resident atomic scatter-add.
// ---------------------------------------------------------------------------
__global__ __launch_bounds__(256) void scatter_edges(
    const float* __restrict__ x, const int* __restrict__ esrc,
    const int* __restrict__ edst, const float* __restrict__ ew,
    float* __restrict__ agg)
{
    size_t gid = (size_t)blockIdx.x * blockDim.x + threadIdx.x;
    size_t e = gid >> 5;
    if (e >= (size_t)GIN_E) return;
    int q = (int)(gid & 31) * 4;
    int s = esrc[e];
    int d = edst[e];
    float w = ew[e];
    const float4 v = *reinterpret_cast<const float4*>(x + (size_t)s * D_IN + q);
    float* dst = agg + (size_t)d * D_IN + q;
    atomicAdd(dst + 0, v.x * w);
    atomicAdd(dst + 1, v.y * w);
    atomicAdd(dst + 2, v.z * w);
    atomicAdd(dst + 3, v.w * w);
}

// ---------------------------------------------------------------------------
// GEMM1: h1[N,512] = ((1+eps)x + agg)[N,128] @ w1[128,512] + b1
// Block = 8 wave32s, 32-row M tile. A tile (32 x 128) staged in LDS
// transposed [k][m] with odd stride 33 -> conflict-free ds reads for both
// lane halves and both M sub-tiles. Each wave owns a 16-col slice; each
// B fragment (2 global loads) feeds two v_wmma_f32_16x16x4_f32.
// ---------------------------------------------------------------------------
__global__ __launch_bounds__(256) void gemm1_wmma(
    const float* __restrict__ x, const float* __restrict__ agg,
    const float* __restrict__ w1, const float* __restrict__ b1,
    float* __restrict__ h1)
{
    __shared__ float aT[KC][33];   // [k][m], m = 0..31

    const int tid  = threadIdx.x;
    const int lane = tid & 31;
    const int wave = tid >> 5;
    const int row0 = blockIdx.x * 32;

    // Cooperative A-tile load: coalesced along k, fused (1+eps)x + agg.
    for (int i = tid; i < 32 * KC; i += 256) {
        int m = i >> 7;          // 0..31
        int k = i & (KC - 1);    // 0..127
        int r = row0 + m;
        float v = 0.0f;
        if (r < GIN_N) {
            size_t off = (size_t)r * D_IN + k;
            v = (1.0f + EPS1) * x[off] + agg[off];
        }
        aT[k][m] = v;
    }
    __syncthreads();

    const int col  = blockIdx.y * 128 + wave * 16 + (lane & 15);
    const int ksel = (lane >> 4) << 1;   // 0 (lanes 0-15) or 2 (lanes 16-31)
    const int mrow = lane & 15;

    v8f c0, c1;
    const float bias = b1[col];
#pragma unroll
    for (int j = 0; j < 8; ++j) { c0[j] = bias; c1[j] = bias; }

#pragma unroll
    for (int k0 = 0; k0 < D_IN; k0 += 4) {
        v2f a0, a1, b;
        a0.x = aT[k0 + ksel][mrow];
        a0.y = aT[k0 + ksel + 1][mrow];
        a1.x = aT[k0 + ksel][mrow + 16];
        a1.y = aT[k0 + ksel + 1][mrow + 16];
        b.x  = w1[(size_t)(k0 + ksel) * D_HID + col];
        b.y  = w1[(size_t)(k0 + ksel + 1) * D_HID + col];
        c0 = __builtin_amdgcn_wmma_f32_16x16x4_f32(
                 false, a0, false, b, (short)0, c0, false, false);
        c1 = __builtin_amdgcn_wmma_f32_16x16x4_f32(
                 false, a1, false, b, (short)0, c1, false, false);
    }

    const int rbase = row0 + ((lane >> 4) << 3);  // +8 rows for upper half
#pragma unroll
    for (int j = 0; j < 8; ++j) {
        int r = rbase + j;
        if (r < GIN_N)      h1[(size_t)r * D_HID + col] = c0[j];
        if (r + 16 < GIN_N) h1[(size_t)(r + 16) * D_HID + col] = c1[j];
    }
}

// ---------------------------------------------------------------------------
// Column sums & sum-of-squares of h1 (for batch-mode BN statistics).
// ---------------------------------------------------------------------------
__global__ __launch_bounds__(256) void col_stats(
    const float* __restrict__ h1, float* __restrict__ sum,
    float* __restrict__ sumsq)
{
    const int col = blockIdx.x * blockDim.x + threadIdx.x;  // 0..511
    const int rows_per = (GIN_N + 511) / 512;               // 196
    int r0 = blockIdx.y * rows_per;
    int r1 = r0 + rows_per;
    if (r1 > GIN_N) r1 = GIN_N;
    float s = 0.0f, ss = 0.0f;
    for (int r = r0; r < r1; ++r) {
        float v = h1[(size_t)r * D_HID + col];
        s  += v;
        ss += v * v;
    }
    atomicAdd(&sum[col], s);
    atomicAdd(&sumsq[col], ss);
}

__global__ __launch_bounds__(256) void bn_finalize(
    const float* __restrict__ sum, const float* __restrict__ sumsq,
    const float* __restrict__ gamma, const float* __restrict__ beta,
    float* __restrict__ scale, float* __restrict__ shift)
{
    int c = blockIdx.x * blockDim.x + threadIdx.x;
    if (c >= D_HID) return;
    const float invN = 1.0f / (float)GIN_N;
    float mean = sum[c] * invN;
    float var  = sumsq[c] * invN - mean * mean;   // biased variance
    float sc   = gamma[c] * rsqrtf(var + BN_EPS);
    scale[c] = sc;
    shift[c] = beta[c] - mean * sc;
}

// ---------------------------------------------------------------------------
// GEMM2: out[N,256] = relu(h1*scale + shift)[N,512] @ w2[512,256] + b2
// BN + ReLU fused into the A-tile load. 32-row M tile, K chunked at 128
// (LDS stays at 16.9 KB; accumulators persist in VGPRs across chunks).
// ---------------------------------------------------------------------------
__global__ __launch_bounds__(256) void gemm2_wmma(
    const float* __restrict__ h1, const float* __restrict__ scale,
    const float* __restrict__ shift, const float* __restrict__ w2,
    const float* __restrict__ b2, float* __restrict__ out)
{
    __shared__ float aT[KC][33];   // [k][m], m = 0..31

    const int tid  = threadIdx.x;
    const int lane = tid & 31;
    const int wave = tid >> 5;
    const int row0 = blockIdx.x * 32;

    const int col  = blockIdx.y * 128 + wave * 16 + (lane & 15);
    const int ksel = (lane >> 4) << 1;
    const int mrow = lane & 15;

    v8f c0, c1;
    const float bias = b2[col];
#pragma unroll
    for (int j = 0; j < 8; ++j) { c0[j] = bias; c1[j] = bias; }

    for (int kb = 0; kb < D_HID; kb += KC) {
        // Stage A chunk with fused BN + ReLU.
        for (int i = tid; i < 32 * KC; i += 256) {
            int m  = i >> 7;          // 0..31
            int kc = i & (KC - 1);    // 0..127
            int r  = row0 + m;
            float v = 0.0f;
            if (r < GIN_N) {
                int k = kb + kc;
                float h = h1[(size_t)r * D_HID + k];
                h = h * scale[k] + shift[k];
                v = h > 0.0f ? h : 0.0f;          // ReLU
            }
            aT[kc][m] = v;
        }
        __syncthreads();

        for (int kc = 0; kc < KC; kc += 4) {
            v2f a0, a1, b;
            a0.x = aT[kc + ksel][mrow];
            a0.y = aT[kc + ksel + 1][mrow];
            a1.x = aT[kc + ksel][mrow + 16];
            a1.y = aT[kc + ksel + 1][mrow + 16];
            int k = kb + kc;
            b.x  = w2[(size_t)(k + ksel) * D_OUT + col];
            b.y  = w2[(size_t)(k + ksel + 1) * D_OUT + col];
            c0 = __builtin_amdgcn_wmma_f32_16x16x4_f32(
                     false, a0, false, b, (short)0, c0, false, false);
            c1 = __builtin_amdgcn_wmma_f32_16x16x4_f32(
                     false, a1, false, b, (short)0, c1, false, false);
        }
        __syncthreads();
    }

    const int rbase = row0 + ((lane >> 4) << 3);
#pragma unroll
    for (int j = 0; j < 8; ++j) {
        int r = rbase + j;
        if (r < GIN_N)      out[(size_t)r * D_OUT + col] = c0[j];
        if (r + 16 < GIN_N) out[(size_t)(r + 16) * D_OUT + col] = c1[j];
    }
}

// ---------------------------------------------------------------------------
extern "C" void kernel_launch(void* const* d_in, const int* in_sizes, int n_in,
                              void* d_out, int out_size, void* d_ws, size_t ws_size,
                              hipStream_t stream) {
    (void)in_sizes; (void)n_in; (void)out_size; (void)ws_size;

    const float* x     = (const float*)d_in[0];
    const int*   esrc  = (const int*)  d_in[1];
    const int*   edst  = (const int*)  d_in[2];
    const float* ew    = (const float*)d_in[3];
    const float* w1    = (const float*)d_in[4];
    const float* b1    = (const float*)d_in[5];
    const float* gamma = (const float*)d_in[6];
    const float* beta  = (const float*)d_in[7];
    const float* w2    = (const float*)d_in[8];
    const float* b2    = (const float*)d_in[9];
    float* out = (float*)d_out;

    float* ws    = (float*)d_ws;
    float* agg   = ws;                                  // N*128
    float* sum   = agg   + WS_AGG_ELEMS;                // 512
    float* sumsq = sum   + D_HID;                       // 512
    float* scale = sumsq + D_HID;                       // 512
    float* shift = scale + D_HID;                       // 512
    float* h1    = shift + D_HID;                       // N*512

    // 1) zero agg + sum + sumsq (contiguous prefix of ws)
    {
        size_t zc = WS_AGG_ELEMS + 2 * (size_t)D_HID;
        zero_f32<<<4096, 256, 0, stream>>>(ws, zc);
    }

    // 2) edge scatter-add: E*32 threads (4 feats each)
    {
        size_t total = (size_t)GIN_E * 32;
        dim3 grid((unsigned)((total + 255) / 256));
        scatter_edges<<<grid, 256, 0, stream>>>(x, esrc, edst, ew, agg);
    }

    // 3) GEMM1 + bias (WMMA f32, 32-row M tiles)
    {
        dim3 grid((GIN_N + 31) / 32, D_HID / 128);   // (3125, 4)
        gemm1_wmma<<<grid, 256, 0, stream>>>(x, agg, w1, b1, h1);
    }

    // 4) BN batch stats
    {
        dim3 grid(D_HID / 256, 512);                 // (2, 512)
        col_stats<<<grid, 256, 0, stream>>>(h1, sum, sumsq);
    }

    // 5) finalize scale/shift
    bn_finalize<<<(D_HID + 255) / 256, 256, 0, stream>>>(sum, sumsq, gamma, beta,
                                                         scale, shift);

    // 6) GEMM2 with fused BN + ReLU on A, bias in accumulator
    {
        dim3 grid((GIN_N + 31) / 32, D_OUT / 128);   // (3125, 2)
        gemm2_wmma<<<grid, 256, 0, stream>>>(h1, scale, shift, w2, b2, out);
    }
}